// sgd_sw_29188597744222
// MI455X (gfx1250) — compile-verified
//
#include <hip/hip_runtime.h>

// Sequential logistic-regression SGD, exactly re-blocked so the dominant math
// is a 16x16xK Gram-tile product on V_WMMA_F32_16X16X4_F32 (fp32 WMMA — the
// chain is latency-bound at ~2.4 GFLOP total, so fp32 fidelity costs nothing).
// One persistent 512-thread workgroup walks the 4096 sequential block steps.
// CDNA5 async path: the next block's 64KB X-tile streams into a double-buffered
// LDS tile via GLOBAL_LOAD_ASYNC_TO_LDS_B128 (ASYNCcnt) while the current block
// computes, hiding L2/HBM latency on the sequential critical path.
// X (134 MB) is L2-resident (192 MB L2), so epoch-2 prefetches are L2 hits.

#define F        1024
#define NTRAIN   32768
#define EPOCHS   2
#define LR       0.01f
#define BLK      16              // samples per sequential block = one WMMA tile
#define NBLOCKS  (NTRAIN / BLK)  // 2048 (power of two)
#define NWAVES   16              // 512 threads, wave32
#define KCHUNK   (F / NWAVES)    // K-range per wave for the Gram product
#define XPITCH   (F + 4)         // LDS row pitch (floats): conflict-free b64 reads

typedef float v2f __attribute__((ext_vector_type(2)));
typedef float v8f __attribute__((ext_vector_type(8)));
typedef int   v4i __attribute__((ext_vector_type(4)));
typedef v4i __attribute__((address_space(1))) gv4i_t;   // global int4
typedef v4i __attribute__((address_space(3))) lv4i_t;   // LDS int4

// Issue the async global->LDS copy of one 16x1024 fp32 tile (4096 B128 moves,
// 8 per thread across 512 threads). Completion tracked by ASYNCcnt.
__device__ __forceinline__ void prefetch_tile_async(const float* __restrict__ gsrc,
                                                    float* ldst, int tid) {
#pragma unroll
    for (int i = 0; i < 8; ++i) {
        const int q    = tid + i * 512;   // 0..4095
        const int row  = q >> 8;          // 16 rows, 256 B128 each
        const int col4 = (q & 255) << 2;  // float offset within row
        __builtin_amdgcn_global_load_async_to_lds_b128(
            (gv4i_t*)(gsrc + row * F + col4),
            (lv4i_t*)(ldst + row * XPITCH + col4),
            /*offset=*/0, /*cpol=*/0);
    }
}

__device__ __forceinline__ void wait_async_all() {
    asm volatile("s_wait_asynccnt 0x0" ::: "memory");
}

__global__ __launch_bounds__(512, 1)
void sgd_logreg_wmma_kernel(const float* __restrict__ X,
                            const float* __restrict__ y,
                            const float* __restrict__ theta0,
                            float* __restrict__ out) {
    __shared__ float s_x[2][BLK * XPITCH];          // double-buffered X tile (~129KB)
    __shared__ float s_theta[F];                    // current theta
    __shared__ float s_part[NWAVES * BLK * BLK];    // per-wave partial Gram tiles
    __shared__ float s_g[BLK * BLK];                // reduced 16x16 Gram tile
    __shared__ float s_u[BLK];                      // x_i . theta_blockstart
    __shared__ float s_c[BLK];                      // per-sample step coefficients

    const int tid  = threadIdx.x;
    const int lane = tid & 31;
    const int wave = tid >> 5;

    // Preload tile for step 0 while theta is copied in.
    prefetch_tile_async(X, &s_x[0][0], tid);
    for (int f = tid; f < F; f += blockDim.x) s_theta[f] = theta0[f];
    wait_async_all();
    __syncthreads();

    int cur = 0;
    const int TOTAL = EPOCHS * NBLOCKS;
    for (int s = 0; s < TOTAL; ++s) {
        const int    base = (s & (NBLOCKS - 1)) * BLK;
        const float* xt   = &s_x[cur][0];

        // ---- (0) kick off async prefetch of the next block's tile
        if (s + 1 < TOTAL) {
            const int nbase = ((s + 1) & (NBLOCKS - 1)) * BLK;
            prefetch_tile_async(X + (size_t)nbase * F, &s_x[cur ^ 1][0], tid);
        }

        // ---- (1) Gram tile G = Xb * Xb^T, K split across 16 waves.
        // fp32 WMMA A-fragment layout (16x4): lanes 0-15 hold row M=lane,
        // VGPR0/1 = k,k+1; lanes 16-31 hold row M=lane-16, k+2,k+3.
        // G is symmetric and B stripes identically => B-frag == A-frag.
        {
            const int    m    = lane & 15;
            const int    koff = (lane >> 4) * 2;
            const float* rowp = xt + m * XPITCH + wave * KCHUNK + koff;
            v8f acc = {0.f, 0.f, 0.f, 0.f, 0.f, 0.f, 0.f, 0.f};
#pragma unroll
            for (int k = 0; k < KCHUNK; k += 4) {
                v2f a;
                a.x = rowp[k];
                a.y = rowp[k + 1];
                acc = __builtin_amdgcn_wmma_f32_16x16x4_f32(
                    false, a, false, a, (short)0, acc, false, false);
            }
            // C layout: VGPR r -> M = r + (lane<16 ? 0 : 8), N = lane&15
            const int mrow = (lane < 16) ? 0 : 8;
#pragma unroll
            for (int r = 0; r < 8; ++r)
                s_part[wave * 256 + (mrow + r) * 16 + (lane & 15)] = acc[r];
        }

        // ---- (2) u_i = x_i . theta_blockstart  (wave w reduces sample w)
        {
            float        acc = 0.f;
            const float* xr  = xt + wave * XPITCH;
#pragma unroll 8
            for (int f2 = lane; f2 < F; f2 += 32) acc += xr[f2] * s_theta[f2];
#pragma unroll
            for (int off = 16; off > 0; off >>= 1)
                acc += __shfl_down(acc, off, 32);
            if (lane == 0) s_u[wave] = acc;
        }
        __syncthreads();

        // reduce the 16 partial Gram tiles
        if (tid < 256) {
            float g = 0.f;
#pragma unroll
            for (int w = 0; w < NWAVES; ++w) g += s_part[w * 256 + tid];
            s_g[tid] = g;
        }
        __syncthreads();

        // ---- (3) exact in-block recursion (wave 0): lane k carries z_k.
        //   z_i = u_i - sum_{j<i} c_j G[j][i];  c_i = LR*(sigmoid(z_i)-y_i)
        if (wave == 0) {
            float z  = (lane < BLK) ? s_u[lane] : 0.f;
            float yl = (lane < BLK) ? y[base + lane] : 0.f;
            float grow[BLK];
#pragma unroll
            for (int i = 0; i < BLK; ++i)
                grow[i] = (lane < BLK) ? s_g[i * 16 + lane] : 0.f;
#pragma unroll
            for (int i = 0; i < BLK; ++i) {
                float zi  = __shfl(z, i, 32);
                float yi  = __shfl(yl, i, 32);
                float sig = 1.0f / (1.0f + __expf(-zi));
                float c   = LR * (sig - yi);
                if (lane == i) s_c[i] = c;
                z -= c * grow[i];   // harmless for k<=i; exact for k>i
            }
        }
        __syncthreads();

        // ---- (4) theta -= Xb^T c  (thread t handles features t, t+512)
        for (int f2 = tid; f2 < F; f2 += blockDim.x) {
            float th = s_theta[f2];
#pragma unroll
            for (int i = 0; i < BLK; ++i)
                th -= s_c[i] * xt[i * XPITCH + f2];
            s_theta[f2] = th;
        }

        // ---- (5) block boundary: next tile must be resident, all stages done
        wait_async_all();
        __syncthreads();
        cur ^= 1;
    }

    for (int f = tid; f < F; f += blockDim.x) out[f] = s_theta[f];
}

extern "C" void kernel_launch(void* const* d_in, const int* in_sizes, int n_in,
                              void* d_out, int out_size, void* d_ws, size_t ws_size,
                              hipStream_t stream) {
    (void)in_sizes; (void)n_in; (void)out_size; (void)d_ws; (void)ws_size;
    const float* data  = (const float*)d_in[0];   // [32768, 1024] f32
    const float* label = (const float*)d_in[1];   // [32768]       f32
    const float* theta = (const float*)d_in[2];   // [1024]        f32
    float*       outp  = (float*)d_out;           // [1024]        f32
    sgd_logreg_wmma_kernel<<<dim3(1), dim3(512), 0, stream>>>(data, label, theta, outp);
}